// HypCD_43155831390572
// MI455X (gfx1250) — compile-verified
//
#include <hip/hip_runtime.h>
#include <math.h>

typedef __attribute__((ext_vector_type(2))) float v2f;
typedef __attribute__((ext_vector_type(8))) float v8f;

#define D_DIM 128

__device__ __forceinline__ float wave_half_reduce(float v) {
    // butterfly across the 16-lane half (xor 8,4,2,1 stays within each half)
    v += __shfl_xor(v, 8, 32);
    v += __shfl_xor(v, 4, 32);
    v += __shfl_xor(v, 2, 32);
    v += __shfl_xor(v, 1, 32);
    return v;
}

// ---------------------------------------------------------------------------
// Kernel 1: squared row norms of Z (into na) and Z' (into nb).
// One wave32 per row; each lane loads a float4 (32*4 = 128 = D).
// ---------------------------------------------------------------------------
__global__ void __launch_bounds__(256) hyp_row_norms(const float* __restrict__ z,
                                                     const float* __restrict__ zp,
                                                     float* __restrict__ na,
                                                     float* __restrict__ nb,
                                                     int B) {
    int gw   = (blockIdx.x * blockDim.x + threadIdx.x) >> 5;  // global wave id
    int lane = threadIdx.x & 31;
    if (gw >= 2 * B) return;  // wave-uniform
    const float* src = (gw < B) ? (z + (size_t)gw * D_DIM)
                                : (zp + (size_t)(gw - B) * D_DIM);
    float4 v = *(const float4*)(src + lane * 4);
    float s = v.x * v.x + v.y * v.y + v.z * v.z + v.w * v.w;
    s += __shfl_xor(s, 16, 32);
    s = wave_half_reduce(s);
    if (lane == 0) {
        if (gw < B) na[gw] = s;
        else        nb[gw - B] = s;
    }
}

// ---------------------------------------------------------------------------
// Kernel 2: fused WMMA GEMM (G = Z @ Z'^T) + hyperbolic/cosine similarity
// epilogue + masked row sums. One workgroup = 16 rows; 8 waves sweep the 64
// column tiles; K-loop uses v_wmma_f32_16x16x4_f32 (exact fp32).
// ---------------------------------------------------------------------------
__global__ void __launch_bounds__(256) hyp_sim_rows(const float* __restrict__ Z,
                                                    const float* __restrict__ Zp,
                                                    const float* __restrict__ na,
                                                    const float* __restrict__ nb,
                                                    float* __restrict__ denom,
                                                    float* __restrict__ pos,
                                                    int B) {
    constexpr float C      = 0.05f;
    constexpr float EPS    = 1e-06f;
    constexpr float INV_T  = 10.0f;       // 1 / TEMPERATURE
    const float sqrt_c     = 0.22360679774997896f;   // sqrt(0.05)
    const float inv_sqrt_c = 4.47213595499958f;      // 1/sqrt(0.05)

    const int row0 = blockIdx.x * 16;
    const int tid  = threadIdx.x;
    const int wave = tid >> 5;
    const int lane = tid & 31;
    const int lo   = lane & 15;
    const int hi   = lane >> 4;

    __shared__ float sRowSum[16];
    __shared__ float sPos[16];
    __shared__ float sNormA[16];
    if (tid < 16) {
        sRowSum[tid] = 0.0f;
        sPos[tid]    = 0.0f;
        sNormA[tid]  = na[row0 + tid];
    }
    __syncthreads();

    // Preload A fragments for this row block (reused across all column tiles).
    // f32 16x4 A layout: lane holds row m=lo; VGPR pair holds K = 4*kk + 2*hi, +1.
    float2 afrag[32];
    const float* Arow = Z + (size_t)(row0 + lo) * D_DIM;
#pragma unroll
    for (int kk = 0; kk < 32; ++kk)
        afrag[kk] = *(const float2*)(Arow + 4 * kk + 2 * hi);

    // per-lane row norms for the 8 accumulator rows (m = r + 8*hi)
    float x2v[8], xnv[8];
#pragma unroll
    for (int r = 0; r < 8; ++r) {
        x2v[r] = sNormA[r + 8 * hi];
        xnv[r] = fmaxf(sqrtf(x2v[r]), 1e-12f);
    }

    float preg[8];
#pragma unroll
    for (int r = 0; r < 8; ++r) preg[r] = 0.0f;

    const int ntiles = B >> 4;
    for (int t = wave; t < ntiles; t += 8) {
        const int col0 = t << 4;

        v8f acc = {0.f, 0.f, 0.f, 0.f, 0.f, 0.f, 0.f, 0.f};
        // B = Z'^T so B[k][n] = Zp[n][k]; lane holds column n=lo, K pair at 4*kk+2*hi
        const float* Brow = Zp + (size_t)(col0 + lo) * D_DIM;
#pragma unroll
        for (int kk = 0; kk < 32; ++kk) {
            float2 b2 = *(const float2*)(Brow + 4 * kk + 2 * hi);
            v2f av = {afrag[kk].x, afrag[kk].y};
            v2f bv = {b2.x, b2.y};
            acc = __builtin_amdgcn_wmma_f32_16x16x4_f32(
                false, av, false, bv, (short)0, acc, false, false);
        }

        // epilogue: per-lane column j = col0 + lo (same for all 8 rows)
        const int   j     = col0 + lo;
        const float y2    = nb[j];
        const float ynorm = fmaxf(sqrtf(y2), 1e-12f);

#pragma unroll
        for (int r = 0; r < 8; ++r) {
            const int   m  = r + 8 * hi;
            const int   i  = row0 + m;
            const float g  = acc[r];          // z_i . z'_j
            const float x2 = x2v[r];

            const float xy  = -g;
            const float u   = 1.0f + 2.0f * C * xy + C * y2;
            const float v   = 1.0f - C * x2;
            const float den = fmaxf(1.0f + 2.0f * C * xy + C * C * x2 * y2, EPS);
            const float m2  = (u * u * x2 + 2.0f * u * v * xy + v * v * y2) / (den * den);
            const float mn  = sqrtf(fmaxf(m2, 0.0f));
            const float tt  = fminf(sqrt_c * mn, 1.0f - EPS);
            // dist = (2/sqrt_c)*atanh(tt) = (1/sqrt_c)*log((1+tt)/(1-tt))
            const float dist = inv_sqrt_c * logf((1.0f + tt) / (1.0f - tt));

            const float dsim = -dist * INV_T;
            const float asim = g / (xnv[r] * ynorm) * INV_T;
            const float comb = 0.5f * dsim + 0.5f * asim;

            if (i == j) {
                sPos[m] = comb;               // single writer per row
            } else {
                preg[r] += expf(comb);
            }
        }
    }

    // reduce per-row partials across the 16 lanes of each half, into LDS
#pragma unroll
    for (int r = 0; r < 8; ++r) {
        float v = wave_half_reduce(preg[r]);
        if (lo == 0) atomicAdd(&sRowSum[r + 8 * hi], v);
    }
    __syncthreads();

    if (tid < 16) {
        denom[row0 + tid] = sRowSum[tid];
        pos[row0 + tid]   = sPos[tid];
    }
}

// ---------------------------------------------------------------------------
// Kernel 3: loss = mean(-pos + log(denom)), single block reduction.
// ---------------------------------------------------------------------------
__global__ void __launch_bounds__(256) hyp_loss_reduce(const float* __restrict__ pos,
                                                       const float* __restrict__ denom,
                                                       float* __restrict__ out,
                                                       int B) {
    __shared__ float s[256];
    const int tid = threadIdx.x;
    float acc = 0.0f;
    for (int i = tid; i < B; i += 256)
        acc += logf(denom[i]) - pos[i];
    s[tid] = acc;
    __syncthreads();
    for (int o = 128; o > 0; o >>= 1) {
        if (tid < o) s[tid] += s[tid + o];
        __syncthreads();
    }
    if (tid == 0) out[0] = s[0] / (float)B;
}

extern "C" void kernel_launch(void* const* d_in, const int* in_sizes, int n_in,
                              void* d_out, int out_size, void* d_ws, size_t ws_size,
                              hipStream_t stream) {
    (void)n_in; (void)out_size; (void)ws_size;
    const float* z  = (const float*)d_in[0];
    const float* zp = (const float*)d_in[1];
    const int B = in_sizes[0] / D_DIM;   // 1024

    float* na    = (float*)d_ws;
    float* nb    = na + B;
    float* denom = nb + B;
    float* pos   = denom + B;

    // 2B waves of 32 threads for row norms
    int norm_threads = 2 * B * 32;
    hyp_row_norms<<<(norm_threads + 255) / 256, 256, 0, stream>>>(z, zp, na, nb, B);
    hyp_sim_rows<<<B / 16, 256, 0, stream>>>(z, zp, na, nb, denom, pos, B);
    hyp_loss_reduce<<<1, 256, 0, stream>>>(pos, denom, (float*)d_out, B);
}